// Net_chan5p1_24713241821270
// MI455X (gfx1250) — compile-verified
//
#include <hip/hip_runtime.h>
#include <hip/hip_bf16.h>
#include <math.h>

// MI455X / gfx1250, wave32. f32 WMMA 16x16x4 for all node-level GEMMs;
// edge scatter/gather phases are HBM/L2-bandwidth bound (xp/agg tiles fit in
// the 192MB L2, so float atomics resolve near-memory).

typedef __attribute__((ext_vector_type(2))) float v2f;
typedef __attribute__((ext_vector_type(8))) float v8f;

#define LEAKY 0.2f
#define EPSBN 1e-5f

__device__ __forceinline__ unsigned fenc(float f) {
  unsigned u = __float_as_uint(f);
  return (u & 0x80000000u) ? ~u : (u | 0x80000000u);
}
__device__ __forceinline__ float fdec(unsigned e) {
  return __uint_as_float((e & 0x80000000u) ? (e ^ 0x80000000u) : ~e);
}
__device__ __forceinline__ void edge_nodes(const int* ei, int E, int e, int& s, int& d) {
  if (e < E) { s = ei[e]; d = ei[E + e]; } else { s = e - E; d = e - E; }
}

__global__ void k_fill(float* p, float v, int n) {
  int i = blockIdx.x * blockDim.x + threadIdx.x;
  if (i < n) p[i] = v;
}

// Y[M,F] = X[M,K] @ W[K,F] (+bias). One wave32 per 16x16 tile, 4 waves/block.
// Both operand tiles staged in LDS, zero-padded to Kp=ceil4(K) so the WMMA
// loop has no masking. A frag (16x4 f32): lane<16 -> K={k0,k0+1},
// lane>=16 -> K={k0+2,k0+3}. M must be a multiple of 16. K <= 192.
#define SX_STRIDE 196  // even (8B-aligned b64 A reads), 196 % 64 != 0
__global__ __launch_bounds__(128) void k_gemm_wmma(
    const float* __restrict__ X, const float* __restrict__ W,
    const float* __restrict__ bias, float* __restrict__ Y,
    int M, int K, int F) {
  __shared__ float sW[192 * 64];
  __shared__ float sX[16 * SX_STRIDE];
  const int F0 = blockIdx.y * 64;
  const int Kp = (K + 3) & ~3;
  // stage W tile [Kp x 64], zero-padded rows/cols (coalesced over columns)
  for (int idx = threadIdx.x; idx < Kp * 64; idx += 128) {
    int k = idx >> 6, c = idx & 63;
    int col = F0 + c;
    sW[idx] = (k < K && col < F) ? W[k * F + col] : 0.f;
  }
  // stage X tile [16 x Kp]: contiguous 16*K floats of X -> fully coalesced
  const int r0 = blockIdx.x * 16;
  for (int idx = threadIdx.x; idx < 16 * Kp; idx += 128) {
    int r = idx / Kp, k = idx - r * Kp;
    sX[r * SX_STRIDE + k] = (k < K) ? X[(size_t)(r0 + r) * K + k] : 0.f;
  }
  __syncthreads();
  const int wv   = threadIdx.x >> 5;
  const int lane = threadIdx.x & 31;
  const int half = lane >> 4;
  const int l16  = lane & 15;
  const int tileF = blockIdx.y * 4 + wv;
  if (tileF * 16 >= F) return;  // wave-uniform: EXEC stays all-ones for WMMA
  const float* xr = sX + l16 * SX_STRIDE + half * 2;
  const float* wr = sW + half * 2 * 64 + wv * 16 + l16;
  v8f acc = {};
  for (int k0 = 0; k0 < Kp; k0 += 4) {
    v2f a, b;
    a.x = xr[k0];         // one b64 LDS read
    a.y = xr[k0 + 1];
    b.x = wr[k0 * 64];
    b.y = wr[k0 * 64 + 64];
    acc = __builtin_amdgcn_wmma_f32_16x16x4_f32(false, a, false, b, (short)0,
                                                acc, false, false);
  }
  const int col = tileF * 16 + l16;
  const float bv = bias ? bias[col] : 0.f;
#pragma unroll
  for (int i = 0; i < 8; ++i) {
    int r = r0 + i + half * 8;               // C layout: VGPR i -> M=i / i+8
    Y[(size_t)r * F + col] = acc[i] + bv;
  }
}

// small dense: thread per output element (MLP branch + final FC)
__global__ void k_dense(const float* __restrict__ X, const float* __restrict__ W,
                        const float* __restrict__ B, float* __restrict__ Y,
                        int M, int K, int F, int relu) {
  int i = blockIdx.x * blockDim.x + threadIdx.x;
  if (i >= M * F) return;
  int m = i / F, f = i - m * F;
  float s = B ? B[f] : 0.f;
  const float* xr = X + (size_t)m * K;
  for (int k = 0; k < K; ++k) s += xr[k] * W[k * F + f];
  if (relu) s = fmaxf(s, 0.f);
  Y[i] = s;
}

// per-node attention logits: asrc[n,h] = sum_c xp[n,h,c] * a_src[h,c]
__global__ void k_alpha(const float* __restrict__ xp, const float* __restrict__ aw_s,
                        const float* __restrict__ aw_d, float* __restrict__ asrc,
                        float* __restrict__ adst, int N, int H, int CH) {
  int i = blockIdx.x * blockDim.x + threadIdx.x;
  if (i >= N * H) return;
  int n = i / H, h = i - n * H;
  const float* xr = xp + (size_t)n * H * CH + h * CH;
  float s1 = 0.f, s2 = 0.f;
  for (int c = 0; c < CH; ++c) {
    float v = xr[c];
    s1 += v * aw_s[h * CH + c];
    s2 += v * aw_d[h * CH + c];
  }
  asrc[i] = s1; adst[i] = s2;
}

__global__ void k_edge_max(const int* __restrict__ ei, int E, int N,
                           const float* __restrict__ asrc, const float* __restrict__ adst,
                           unsigned* __restrict__ menc, int H) {
  int t = blockIdx.x * blockDim.x + threadIdx.x;
  if (t >= (E + N) * H) return;
  int e = t / H, h = t - e * H;
  int s, d; edge_nodes(ei, E, e, s, d);
  float al = asrc[s * H + h] + adst[d * H + h];
  al = al > 0.f ? al : LEAKY * al;
  atomicMax(&menc[d * H + h], fenc(al));
}

__global__ void k_edge_expsum(const int* __restrict__ ei, int E, int N,
                              const float* __restrict__ asrc, const float* __restrict__ adst,
                              const unsigned* __restrict__ menc, float* __restrict__ ssum,
                              int H) {
  int t = blockIdx.x * blockDim.x + threadIdx.x;
  if (t >= (E + N) * H) return;
  int e = t / H, h = t - e * H;
  int s, d; edge_nodes(ei, E, e, s, d);
  float al = asrc[s * H + h] + adst[d * H + h];
  al = al > 0.f ? al : LEAKY * al;
  atomicAdd(&ssum[d * H + h], expf(al - fdec(menc[d * H + h])));
}

// wave32 per edge, lane-per-channel scatter (768B contiguous per edge)
__global__ void k_edge_agg(const int* __restrict__ ei, int E, int N,
                           const float* __restrict__ asrc, const float* __restrict__ adst,
                           const unsigned* __restrict__ menc, const float* __restrict__ ssum,
                           const float* __restrict__ xp, float* __restrict__ out,
                           int H, int CH) {
  int gw = (blockIdx.x * blockDim.x + threadIdx.x) >> 5;
  int lane = threadIdx.x & 31;
  if (gw >= E + N) return;
  int s, d; edge_nodes(ei, E, gw, s, d);
  int HC = H * CH;
  __builtin_prefetch(&xp[(size_t)s * HC], 0, 3);  // global_prefetch_b8
  for (int c = lane; c < HC; c += 32) {
    int h = c / CH;
    float al = asrc[s * H + h] + adst[d * H + h];
    al = al > 0.f ? al : LEAKY * al;
    float w = expf(al - fdec(menc[d * H + h])) / (ssum[d * H + h] + 1e-16f);
    atomicAdd(&out[(size_t)d * HC + c], w * xp[(size_t)s * HC + c]);
  }
}

__global__ void k_deg(const int* __restrict__ ei, int E, int N, float* __restrict__ deg) {
  int e = blockIdx.x * blockDim.x + threadIdx.x;
  if (e >= E + N) return;
  int s, d; edge_nodes(ei, E, e, s, d);
  atomicAdd(&deg[d], 1.f);
}

__global__ void k_gcn_agg(const int* __restrict__ ei, int E, int N,
                          const float* __restrict__ deg, const float* __restrict__ xw,
                          float* __restrict__ out) {
  int gw = (blockIdx.x * blockDim.x + threadIdx.x) >> 5;
  int lane = threadIdx.x & 31;
  if (gw >= E + N) return;
  int s, d; edge_nodes(ei, E, gw, s, d);
  float coef = rsqrtf(fmaxf(deg[s], 1.f)) * rsqrtf(fmaxf(deg[d], 1.f));
  atomicAdd(&out[(size_t)d * 32 + lane], coef * xw[(size_t)s * 32 + lane]);
}

__global__ void k_bias_act(float* __restrict__ h, const float* __restrict__ b,
                           int n, int F, int relu) {
  int i = blockIdx.x * blockDim.x + threadIdx.x;
  if (i >= n) return;
  float v = h[i] + b[i % F];
  if (relu) v = fmaxf(v, 0.f);
  h[i] = v;
}

// blockDim == F; thread f owns column f over a chunk of rows (coalesced)
__global__ void k_bn_stats(const float* __restrict__ h, int Nrows, int F,
                           int rows_per_blk, float* __restrict__ st) {
  int f = threadIdx.x;
  int r0 = blockIdx.x * rows_per_blk;
  int r1 = r0 + rows_per_blk; if (r1 > Nrows) r1 = Nrows;
  float s = 0.f, q = 0.f;
  for (int r = r0; r < r1; ++r) { float v = h[(size_t)r * F + f]; s += v; q += v * v; }
  atomicAdd(&st[f], s);
  atomicAdd(&st[F + f], q);
}

__global__ void k_bn_apply(const float* __restrict__ h, float* __restrict__ y,
                           int Nrows, int F, const float* __restrict__ st,
                           const float* __restrict__ g, const float* __restrict__ b,
                           const float* __restrict__ addp, int relu_out) {
  int i = blockIdx.x * blockDim.x + threadIdx.x;
  if (i >= Nrows * F) return;
  int f = i % F;
  float mu = st[f] / (float)Nrows;
  float var = st[F + f] / (float)Nrows - mu * mu;   // biased var, matches jnp.var
  float v = (h[i] - mu) * rsqrtf(var + EPSBN) * g[f] + b[f];
  if (addp) v += addp[i];
  if (relu_out) v = fmaxf(v, 0.f);
  y[i] = v;
}

__global__ void k_count(const int* __restrict__ batch, int N, float* __restrict__ cnt) {
  int n = blockIdx.x * blockDim.x + threadIdx.x;
  if (n < N) atomicAdd(&cnt[batch[n]], 1.f);
}

__global__ void k_pool(const int* __restrict__ batch, const float* __restrict__ h,
                       int N, float* __restrict__ pool) {
  int i = blockIdx.x * blockDim.x + threadIdx.x;
  if (i >= N * 64) return;
  int n = i >> 6, c = i & 63;
  atomicAdd(&pool[(size_t)batch[n] * 64 + c], h[i]);
}

__global__ void k_fcin(const float* __restrict__ pool, const float* __restrict__ cnt,
                       const float* __restrict__ gbr, float* __restrict__ fcin, int G) {
  int i = blockIdx.x * blockDim.x + threadIdx.x;
  if (i >= G * 96) return;
  int g = i / 96, j = i - g * 96;
  fcin[i] = (j < 64) ? pool[g * 64 + j] / fmaxf(cnt[g], 1.f)
                     : gbr[g * 32 + (j - 64)];
}

extern "C" void kernel_launch(void* const* d_in, const int* in_sizes, int n_in,
                              void* d_out, int out_size, void* d_ws, size_t ws_size,
                              hipStream_t stream) {
  const float* x      = (const float*)d_in[0];
  const int*   ei     = (const int*)d_in[1];
  const int*   batch  = (const int*)d_in[3];
  const float* glcm   = (const float*)d_in[4];
  const float* mlp1_w = (const float*)d_in[6];   const float* mlp1_b = (const float*)d_in[7];
  const float* mlp2_w = (const float*)d_in[8];   const float* mlp2_b = (const float*)d_in[9];
  const float* mlp3_w = (const float*)d_in[10];  const float* mlp3_b = (const float*)d_in[11];
  const float* g1_w   = (const float*)d_in[12];
  const float* g1_as  = (const float*)d_in[13];  const float* g1_ad  = (const float*)d_in[14];
  const float* g1_b   = (const float*)d_in[15];
  const float* g1_bn1g= (const float*)d_in[16];  const float* g1_bn1b= (const float*)d_in[17];
  const float* g1_lw  = (const float*)d_in[18];  const float* g1_lb  = (const float*)d_in[19];
  const float* g1_bn2g= (const float*)d_in[20];  const float* g1_bn2b= (const float*)d_in[21];
  const float* gcn_w  = (const float*)d_in[22];  const float* gcn_b  = (const float*)d_in[23];
  const float* gcn_bng= (const float*)d_in[24];  const float* gcn_bnb= (const float*)d_in[25];
  const float* res_w  = (const float*)d_in[26];  const float* res_b  = (const float*)d_in[27];
  const float* g2_w   = (const float*)d_in[28];
  const float* g2_as  = (const float*)d_in[29];  const float* g2_ad  = (const float*)d_in[30];
  const float* g2_b   = (const float*)d_in[31];
  const float* g2_bn1g= (const float*)d_in[32];  const float* g2_bn1b= (const float*)d_in[33];
  const float* g2_lw  = (const float*)d_in[34];  const float* g2_lb  = (const float*)d_in[35];
  const float* g2_bn2g= (const float*)d_in[36];  const float* g2_bn2b= (const float*)d_in[37];
  const float* fc_w   = (const float*)d_in[38];  const float* fc_b   = (const float*)d_in[39];
  float* out = (float*)d_out;

  const int N = in_sizes[0] / 14;
  const int E = in_sizes[1] / 2;
  const int G = in_sizes[4] / 96;
  const int Etot = E + N;

  // workspace layout (f32)
  float* ws = (float*)d_ws;
  size_t o = 0;
  auto alloc = [&](size_t n) { float* p = ws + o; o += n; return p; };
  float* XP   = alloc((size_t)N * 192);   // transformed feats / GCN xw
  float* AGG  = alloc((size_t)N * 192);   // scatter accumulator
  float* H1   = alloc((size_t)N * 16);
  float* H2   = alloc((size_t)N * 32);
  float* H3   = alloc((size_t)N * 64);
  float* RES  = alloc((size_t)N * 32);
  float* ASRC = alloc((size_t)N * 12);
  float* ADST = alloc((size_t)N * 12);
  float* MB   = alloc((size_t)N * 12);    // encoded-uint max buffer
  float* SB   = alloc((size_t)N * 12);
  float* DEG  = alloc((size_t)N);
  float* ST   = alloc(2 * 192);
  float* POOL = alloc((size_t)G * 64);
  float* CNT  = alloc((size_t)G);
  float* GB1  = alloc((size_t)G * 64);
  float* GB2  = alloc((size_t)G * 64);
  float* GBR  = alloc((size_t)G * 32);
  float* FCIN = alloc((size_t)G * 96);
  unsigned* Mu = (unsigned*)MB;

  auto cdiv = [](int a, int b) { return (a + b - 1) / b; };
  const int Mt = N / 16;  // N = 50000 -> 3125 row tiles

  auto fill = [&](float* p, float v, int n) {
    k_fill<<<cdiv(n, 256), 256, 0, stream>>>(p, v, n);
  };
  auto gemm = [&](const float* X, const float* W, const float* B, float* Y, int K, int F) {
    k_gemm_wmma<<<dim3(Mt, cdiv(F, 64)), 128, 0, stream>>>(X, W, B, Y, N, K, F);
  };
  auto bn = [&](const float* in, float* outp, int F, const float* g, const float* b,
                const float* addp, int relu_out) {
    fill(ST, 0.f, 2 * F);
    k_bn_stats<<<cdiv(N, 128), F, 0, stream>>>(in, N, F, 128, ST);
    k_bn_apply<<<cdiv(N * F, 256), 256, 0, stream>>>(in, outp, N, F, ST, g, b, addp, relu_out);
  };
  auto gat_soft = [&](const float* xp, const float* aws, const float* awd,
                      float* aggp, int H, int CH) {
    int HC = H * CH;
    k_alpha<<<cdiv(N * H, 256), 256, 0, stream>>>(xp, aws, awd, ASRC, ADST, N, H, CH);
    fill(MB, 0.f, N * H);   // 0u == encoded -max
    k_edge_max<<<cdiv(Etot * H, 256), 256, 0, stream>>>(ei, E, N, ASRC, ADST, Mu, H);
    fill(SB, 0.f, N * H);
    k_edge_expsum<<<cdiv(Etot * H, 256), 256, 0, stream>>>(ei, E, N, ASRC, ADST, Mu, SB, H);
    fill(aggp, 0.f, N * HC);
    k_edge_agg<<<cdiv(Etot * 32, 256), 256, 0, stream>>>(ei, E, N, ASRC, ADST, Mu, SB,
                                                         xp, aggp, H, CH);
  };

  // ---- GLCM MLP branch: 96 -> 64 -> 64 -> 32
  k_dense<<<cdiv(G * 64, 256), 256, 0, stream>>>(glcm, mlp1_w, mlp1_b, GB1, G, 96, 64, 1);
  k_dense<<<cdiv(G * 64, 256), 256, 0, stream>>>(GB1, mlp2_w, mlp2_b, GB2, G, 64, 64, 1);
  k_dense<<<cdiv(G * 32, 256), 256, 0, stream>>>(GB2, mlp3_w, mlp3_b, GBR, G, 64, 32, 0);

  // ---- GAT block 1 (H=12, CH=16)
  gemm(x, g1_w, nullptr, XP, 14, 192);
  gat_soft(XP, g1_as, g1_ad, AGG, 12, 16);
  k_bias_act<<<cdiv(N * 192, 256), 256, 0, stream>>>(AGG, g1_b, N * 192, 192, 0);
  bn(AGG, AGG, 192, g1_bn1g, g1_bn1b, nullptr, 1);           // relu(bn(h))
  gemm(AGG, g1_lw, g1_lb, H1, 192, 16);
  bn(H1, H1, 16, g1_bn2g, g1_bn2b, nullptr, 1);

  // ---- GCN block 16 -> 32 with residual
  fill(DEG, 0.f, N);
  k_deg<<<cdiv(Etot, 256), 256, 0, stream>>>(ei, E, N, DEG);
  gemm(H1, gcn_w, nullptr, XP, 16, 32);                      // xw (reuses XP)
  fill(AGG, 0.f, N * 32);
  k_gcn_agg<<<cdiv(Etot * 32, 256), 256, 0, stream>>>(ei, E, N, DEG, XP, AGG);
  k_bias_act<<<cdiv(N * 32, 256), 256, 0, stream>>>(AGG, gcn_b, N * 32, 32, 1); // relu(h+b)
  gemm(H1, res_w, res_b, RES, 16, 32);
  bn(AGG, H2, 32, gcn_bng, gcn_bnb, RES, 0);                 // bn(relu(h)) + res

  // ---- GAT block 2 (H=3, CH=64)
  gemm(H2, g2_w, nullptr, XP, 32, 192);
  gat_soft(XP, g2_as, g2_ad, AGG, 3, 64);
  k_bias_act<<<cdiv(N * 192, 256), 256, 0, stream>>>(AGG, g2_b, N * 192, 192, 0);
  bn(AGG, AGG, 192, g2_bn1g, g2_bn1b, nullptr, 1);
  gemm(AGG, g2_lw, g2_lb, H3, 192, 64);
  bn(H3, H3, 64, g2_bn2g, g2_bn2b, nullptr, 1);

  // ---- mean pool + concat + FC head
  fill(POOL, 0.f, G * 64);
  fill(CNT, 0.f, G);
  k_count<<<cdiv(N, 256), 256, 0, stream>>>(batch, N, CNT);
  k_pool<<<cdiv(N * 64, 256), 256, 0, stream>>>(batch, H3, N, POOL);
  k_fcin<<<cdiv(G * 96, 256), 256, 0, stream>>>(POOL, CNT, GBR, FCIN, G);
  k_dense<<<cdiv(G * 10, 256), 256, 0, stream>>>(FCIN, fc_w, fc_b, out, G, 96, 10, 0);
}